// ScRRAMBLeCapsLayerNoPad_1855425872248
// MI455X (gfx1250) — compile-verified
//
#include <hip/hip_runtime.h>

typedef float v2f __attribute__((ext_vector_type(2)));
typedef float v8f __attribute__((ext_vector_type(8)));

#define NUM_CAPS 512
#define CAP_SIZE 256
#define RF_SIZE  64
#define KDIM     2048   // IN_CAPS * RF_PER_CAP  (contraction S)
#define MDIM     2048   // NUM_CAPS * RF_PER_CAP (output rows R)
#define NDIM     64     // RF_SIZE columns
#define KSPLIT   8      // waves per block cooperating on one tile (stage 1)

// ---------------------------------------------------------------------------
// Stage 1: x_routed[R, m] = sum_S Ci_flat[S*2048 + R] * x[S*64 + m]
// GEMM M=2048 N=64 K=2048 via V_WMMA_F32_16X16X4_F32 (full f32 precision).
// One BLOCK per 16x16 output tile; 8 waves split K (256 each) and reduce
// their D fragments through LDS in a fixed order (deterministic, no atomics).
// The K-chunk id is routed through readfirstlane so the loop bound lives in
// an SGPR: the loop compiles to a scalar branch and EXEC stays all-ones
// through every WMMA (an ISA requirement) by construction.
// ---------------------------------------------------------------------------
__global__ __launch_bounds__(256)
void scrramble_route_wmma(const float* __restrict__ Ci,
                          const float* __restrict__ x,
                          float* __restrict__ xr) {
  __shared__ float red[KSPLIT * 32 * 8];   // 8KB of partial D fragments

  const int lane = threadIdx.x & 31;
  // wave-uniform K-chunk id, forced into an SGPR
  const int w    = __builtin_amdgcn_readfirstlane(threadIdx.x >> 5); // 0..7
  const int mt   = blockIdx.x >> 2;        // 0..127
  const int nt   = blockIdx.x & 3;         // 0..3
  const int R0   = mt * 16;
  const int N0   = nt * 16;
  const int half = lane >> 4;              // 0: K,K+1   1: K+2,K+3
  const int ll   = lane & 15;

  const int Sbeg = w * (KDIM / KSPLIT);
  const int Send = Sbeg + (KDIM / KSPLIT);

  v8f c = {};
#pragma unroll 4
  for (int S = Sbeg; S < Send; S += 4) {
    const int k0 = S + 2 * half;
    v2f a, b;
    // A[M=R0+ll, K=k0/k0+1]: adjacent lanes -> adjacent Ci addresses
    a.x = Ci[(size_t)k0 * MDIM + (size_t)(R0 + ll)];
    a.y = Ci[(size_t)(k0 + 1) * MDIM + (size_t)(R0 + ll)];
    // B[K=k0/k0+1, N=N0+ll]: x is 512KB, L2/L0 resident
    b.x = x[k0 * NDIM + N0 + ll];
    b.y = x[(k0 + 1) * NDIM + N0 + ll];
    c = __builtin_amdgcn_wmma_f32_16x16x4_f32(false, a, false, b,
                                              (short)0, c, false, false);
  }

  // Publish this wave's partial D fragment.
#pragma unroll
  for (int r = 0; r < 8; ++r) red[(w * 32 + lane) * 8 + r] = c[r];
  __syncthreads();

  // Wave 0: fixed-order reduction over the 8 K-chunks, then store D.
  if (threadIdx.x < 32) {
    v8f acc = {};
#pragma unroll
    for (int ww = 0; ww < KSPLIT; ++ww) {
#pragma unroll
      for (int r = 0; r < 8; ++r) acc[r] += red[(ww * 32 + lane) * 8 + r];
    }
    // D layout: VGPR r -> row R0 + r + 8*half, col N0 + ll
#pragma unroll
    for (int r = 0; r < 8; ++r)
      xr[(size_t)(R0 + r + 8 * half) * NDIM + (size_t)(N0 + ll)] = acc[r];
  }
}

// ---------------------------------------------------------------------------
// Stage 2: y[cap, row] = sum_col Wcap[row, col] * xr[cap, col]
// row = (j,l): addr = j*16384 + l*64 ; col = (k,m): addr = k*4096 + m.
// Batched mat-vec: vector occupies B column 0 so the 134MB Wi stream feeds
// the WMMA A port as 8B/lane b64 NON-TEMPORAL loads (single-use weights must
// not thrash L2). xr_cap (1KB) is staged in LDS once per block and the
// B-vector entries come from ds_load; a uniform global_prefetch_b8 runs one
// 4KB k-block ahead of the NT stream. One block per capsule, 16 waves.
// ---------------------------------------------------------------------------
__global__ __launch_bounds__(512)
void caps_transform_wmma(const float* __restrict__ Wi,
                         const float* __restrict__ xr,
                         float* __restrict__ y) {
  __shared__ float xs[CAP_SIZE];

  const int cap   = blockIdx.x;          // 0..511
  const int mtile = threadIdx.x >> 5;    // 0..15
  const int lane  = threadIdx.x & 31;
  const int half  = lane >> 4;
  const int ll    = lane & 15;
  const int row   = mtile * 16 + ll;     // 0..255 (16 | 64 -> single j block)
  const int jj    = row >> 6;
  const int l     = row & 63;

  const float* __restrict__ Wcap = Wi + (size_t)cap * (CAP_SIZE * CAP_SIZE);
  const float* __restrict__ Wrow = Wcap + jj * 16384 + l * 64;

  // Stage the capsule's routed vector into LDS (one b32 per thread).
  if (threadIdx.x < CAP_SIZE)
    xs[threadIdx.x] = xr[(size_t)cap * CAP_SIZE + threadIdx.x];
  __syncthreads();

  v8f c = {};
#pragma unroll 4
  for (int S = 0; S < CAP_SIZE; S += 4) {
    const int col0 = S + 2 * half;       // even -> col0,col0+1 in same k block
    const int k    = col0 >> 6;
    const int m    = col0 & 63;
    if ((S & 63) == 0 && S + 64 < CAP_SIZE) {
      // pull the next 4KB k-block of this row group toward the caches
      __builtin_prefetch(Wrow + (k + 1) * 4096 + m, 0, 1);
    }
    // A[row, col0/col0+1]: two consecutive floats -> one b64 NT load
    const v2f a = __builtin_nontemporal_load(
        (const v2f*)(Wrow + k * 4096 + m));
    const float bx0 = xs[col0];
    const float bx1 = xs[col0 + 1];
    v2f b;
    b.x = (ll == 0) ? bx0 : 0.0f;        // vector lives in B column N=0
    b.y = (ll == 0) ? bx1 : 0.0f;
    c = __builtin_amdgcn_wmma_f32_16x16x4_f32(false, a, false, b,
                                              (short)0, c, false, false);
  }

  if (ll == 0) {                         // only the N=0 column is meaningful
#pragma unroll
    for (int r = 0; r < 8; ++r) {
      y[(size_t)cap * CAP_SIZE + (size_t)(mtile * 16 + r + 8 * half)] = c[r];
    }
  }
}

// ---------------------------------------------------------------------------
// Launch: inputs in setup_inputs() order {x, Ci, Wi}; d_ws holds the 512KB
// x_routed intermediate (fully overwritten each call -> deterministic).
// ---------------------------------------------------------------------------
extern "C" void kernel_launch(void* const* d_in, const int* in_sizes, int n_in,
                              void* d_out, int out_size, void* d_ws, size_t ws_size,
                              hipStream_t stream) {
  const float* x  = (const float*)d_in[0];   // 131072
  const float* Ci = (const float*)d_in[1];   // 512*4*512*4
  const float* Wi = (const float*)d_in[2];   // 512*4*4*64*64
  float* y  = (float*)d_out;                 // 512*4*64
  float* xr = (float*)d_ws;                  // 2048*64 floats = 512 KB scratch

  // Stage 1: routing GEMM (M=2048, N=64, K=2048), one block per tile,
  // 8-way in-block K-split.
  scrramble_route_wmma<<<512, 256, 0, stream>>>(Ci, x, xr);
  // Stage 2: 512 capsules x 16 M-tiles, one block per capsule.
  caps_transform_wmma<<<NUM_CAPS, 512, 0, stream>>>(Wi, xr, y);
}